// BiasAddLayerNormFP8Fused_20925080666563
// MI455X (gfx1250) — compile-verified
//
#include <hip/hip_runtime.h>
#include <stdint.h>

// ---------------------------------------------------------------------------
// Fused BiasAdd + Residual + LayerNorm + FP8(E4M3) quant round-trip + amax
// MI455X (gfx1250, wave32). Bandwidth-bound: ~16B HBM traffic / element.
// One wave per row (D=1024 -> 8 x float4 per lane, whole row in VGPRs).
// ---------------------------------------------------------------------------

typedef __attribute__((ext_vector_type(4))) float f4;

#define HIDDEN_D        1024
#define WAVE_SZ         32
#define WAVES_PER_BLOCK 8
#define BLOCK_THREADS   (WAVE_SZ * WAVES_PER_BLOCK)      // 256
#define CHUNKS          (HIDDEN_D / (WAVE_SZ * 4))       // 8 float4 per lane
#define LN_EPS          1e-5f

// ---- wave32 reductions via ds_swizzle (group-of-32 xor mode) --------------
// offset encoding: and_mask=[4:0]=0x1f, or_mask=[9:5]=0, xor_mask=[14:10]
template <int IMM>
__device__ __forceinline__ float swz_xor(float v) {
    return __uint_as_float(
        (unsigned)__builtin_amdgcn_ds_swizzle((int)__float_as_uint(v), IMM));
}
__device__ __forceinline__ float wave_reduce_sum(float v) {
    v += swz_xor<0x401F>(v);  // xor 16
    v += swz_xor<0x201F>(v);  // xor 8
    v += swz_xor<0x101F>(v);  // xor 4
    v += swz_xor<0x081F>(v);  // xor 2
    v += swz_xor<0x041F>(v);  // xor 1
    return v;
}
__device__ __forceinline__ float wave_reduce_max(float v) {
    v = fmaxf(v, swz_xor<0x401F>(v));
    v = fmaxf(v, swz_xor<0x201F>(v));
    v = fmaxf(v, swz_xor<0x101F>(v));
    v = fmaxf(v, swz_xor<0x081F>(v));
    v = fmaxf(v, swz_xor<0x041F>(v));
    return v;
}

// ---- FP8 E4M3 quantize -> dequantize round-trip (hardware converts) -------
__device__ __forceinline__ f4 fp8_roundtrip4(f4 y) {
    int p01 = __builtin_amdgcn_cvt_pk_fp8_f32(y.x, y.y, 0, false);
    int p23 = __builtin_amdgcn_cvt_pk_fp8_f32(y.z, y.w, 0, false);
    f4 r;
    r.x = __builtin_amdgcn_cvt_f32_fp8(p01, 0);
    r.y = __builtin_amdgcn_cvt_f32_fp8(p01, 1);
    r.z = __builtin_amdgcn_cvt_f32_fp8(p23, 0);
    r.w = __builtin_amdgcn_cvt_f32_fp8(p23, 1);
    return r;
}

// ---- amax slot init (harness poisons d_out with 0xAA; uint atomicMax needs 0)
__global__ void ln_amax_init_kernel(float* amax) { *amax = 0.0f; }

__global__ __launch_bounds__(BLOCK_THREADS)
void BiasAddLayerNormFP8Fused_kernel(
    const float* __restrict__ x,
    const float* __restrict__ bias,
    const float* __restrict__ residual,
    const float* __restrict__ gamma,
    const float* __restrict__ beta,
    float* __restrict__ bda_out,
    float* __restrict__ ln_out,
    float* __restrict__ amax_out,
    int n_rows)
{
    __shared__ f4    s_bias [HIDDEN_D / 4];
    __shared__ f4    s_gamma[HIDDEN_D / 4];
    __shared__ f4    s_beta [HIDDEN_D / 4];
    __shared__ float s_wmax [WAVES_PER_BLOCK];

    const int tid  = threadIdx.x;
    const int lane = tid & (WAVE_SZ - 1);
    const int wave = tid >> 5;

    // -- CDNA5 async path: stage the 3 per-column vectors (4KB each) to LDS --
    // Each of the 256 threads async-copies one 16B quad per array (GVS mode:
    // saddr = base pointer, vaddr = 32-bit byte offset, vdst = LDS byte addr).
    {
        unsigned goff  = (unsigned)tid * 16u;
        unsigned lds_b = (unsigned)(uintptr_t)(&s_bias [tid]);
        unsigned lds_g = (unsigned)(uintptr_t)(&s_gamma[tid]);
        unsigned lds_w = (unsigned)(uintptr_t)(&s_beta [tid]);
        asm volatile("global_load_async_to_lds_b128 %0, %1, %2"
                     :: "v"(lds_b), "v"(goff), "s"(bias)  : "memory");
        asm volatile("global_load_async_to_lds_b128 %0, %1, %2"
                     :: "v"(lds_g), "v"(goff), "s"(gamma) : "memory");
        asm volatile("global_load_async_to_lds_b128 %0, %1, %2"
                     :: "v"(lds_w), "v"(goff), "s"(beta)  : "memory");
        asm volatile("s_wait_asynccnt 0" ::: "memory");
    }
    __syncthreads();

    const int rows_per_grid = gridDim.x * WAVES_PER_BLOCK;
    float thread_amax = 0.0f;

    for (int row = blockIdx.x * WAVES_PER_BLOCK + wave; row < n_rows;
         row += rows_per_grid)
    {
        const f4* __restrict__ xr = (const f4*)(x        + (size_t)row * HIDDEN_D);
        const f4* __restrict__ rr = (const f4*)(residual + (size_t)row * HIDDEN_D);
        f4* __restrict__ br = (f4*)(bda_out + (size_t)row * HIDDEN_D);
        f4* __restrict__ lo = (f4*)(ln_out  + (size_t)row * HIDDEN_D);

        // Pass 1: v = x + bias + residual; stream out bda; keep row in VGPRs.
        f4 v[CHUNKS];
        float sum = 0.0f, sumsq = 0.0f;
#pragma unroll
        for (int c = 0; c < CHUNKS; ++c) {
            const int idx = c * WAVE_SZ + lane;
            f4 xv = __builtin_nontemporal_load(&xr[idx]);   // streamed once
            f4 rv = __builtin_nontemporal_load(&rr[idx]);   // streamed once
            f4 bv = s_bias[idx];                            // LDS broadcast
            f4 t;
            t.x = xv.x + bv.x + rv.x;
            t.y = xv.y + bv.y + rv.y;
            t.z = xv.z + bv.z + rv.z;
            t.w = xv.w + bv.w + rv.w;
            v[c] = t;
            __builtin_nontemporal_store(t, &br[idx]);       // don't pollute L2
            sum  += t.x + t.y + t.z + t.w;
            sumsq = fmaf(t.x, t.x, sumsq);
            sumsq = fmaf(t.y, t.y, sumsq);
            sumsq = fmaf(t.z, t.z, sumsq);
            sumsq = fmaf(t.w, t.w, sumsq);
        }

        sum   = wave_reduce_sum(sum);
        sumsq = wave_reduce_sum(sumsq);

        const float inv_d  = 1.0f / (float)HIDDEN_D;
        const float mean   = sum * inv_d;
        const float var    = fmaf(sumsq, inv_d, -mean * mean);
        const float rsigma = rsqrtf(var + LN_EPS);

        // Pass 2 (registers only): normalize, affine, amax, FP8 round-trip.
#pragma unroll
        for (int c = 0; c < CHUNKS; ++c) {
            const int idx = c * WAVE_SZ + lane;
            f4 g  = s_gamma[idx];
            f4 b2 = s_beta [idx];
            f4 t  = v[c];
            f4 y;
            y.x = fmaf((t.x - mean) * rsigma, g.x, b2.x);
            y.y = fmaf((t.y - mean) * rsigma, g.y, b2.y);
            y.z = fmaf((t.z - mean) * rsigma, g.z, b2.z);
            y.w = fmaf((t.w - mean) * rsigma, g.w, b2.w);
            thread_amax = fmaxf(thread_amax, fabsf(y.x));
            thread_amax = fmaxf(thread_amax, fabsf(y.y));
            thread_amax = fmaxf(thread_amax, fabsf(y.z));
            thread_amax = fmaxf(thread_amax, fabsf(y.w));
            __builtin_nontemporal_store(fp8_roundtrip4(y), &lo[idx]);
        }
    }

    // Block-level amax -> single atomic per block (bit-pattern max, amax>=0).
    float wmax = wave_reduce_max(thread_amax);
    if (lane == 0) s_wmax[wave] = wmax;
    __syncthreads();
    if (tid == 0) {
        float bmax = s_wmax[0];
#pragma unroll
        for (int w = 1; w < WAVES_PER_BLOCK; ++w) bmax = fmaxf(bmax, s_wmax[w]);
        atomicMax((unsigned int*)amax_out, __float_as_uint(bmax));
    }
}

extern "C" void kernel_launch(void* const* d_in, const int* in_sizes, int n_in,
                              void* d_out, int out_size, void* d_ws, size_t ws_size,
                              hipStream_t stream) {
    const float* x        = (const float*)d_in[0];
    const float* bias     = (const float*)d_in[1];
    const float* residual = (const float*)d_in[2];
    const float* gamma    = (const float*)d_in[3];
    const float* beta     = (const float*)d_in[4];

    const int d      = in_sizes[1];                    // 1024
    const int n_rows = in_sizes[0] / d;                // 65536

    float* bda  = (float*)d_out;
    float* lno  = bda + (size_t)n_rows * d;
    float* amax = lno + (size_t)n_rows * d;

    ln_amax_init_kernel<<<1, 1, 0, stream>>>(amax);

    const int blocks = (n_rows + WAVES_PER_BLOCK - 1) / WAVES_PER_BLOCK; // 8192
    BiasAddLayerNormFP8Fused_kernel<<<blocks, BLOCK_THREADS, 0, stream>>>(
        x, bias, residual, gamma, beta, bda, lno, amax, n_rows);
}